// GraphAttn_45183055953957
// MI455X (gfx1250) — compile-verified
//
#include <hip/hip_runtime.h>

#define N_NODES 8192
#define F_IN 512
#define F_OUT 512
#define ALPHA 0.2f

typedef __attribute__((ext_vector_type(2))) float v2f;
typedef __attribute__((ext_vector_type(4))) float v4f;
typedef __attribute__((ext_vector_type(8))) float v8f;

// Only check for the WMMA builtin on the DEVICE compile pass; the host pass
// (x86 clang) never declares AMDGCN builtins and must not trip this.
#if defined(__HIP_DEVICE_COMPILE__) && !__has_builtin(__builtin_amdgcn_wmma_f32_16x16x4_f32)
#error "missing __builtin_amdgcn_wmma_f32_16x16x4_f32 on this toolchain (device pass)"
#endif

// ---------------------------------------------------------------------------
// K1: fold W through the attention vector:  w1 = W^T a[:512], w2 = W^T a[512:]
// Packed into a 512x16 fp32 B-matrix (col0 = w1, col1 = w2, cols 2..15 = 0)
// so the score computation becomes a WMMA-shaped GEMM.
// One block of 512 threads; thread c owns column c of W (coalesced row reads).
// ---------------------------------------------------------------------------
__global__ __launch_bounds__(512) void gat_wpack(const float* __restrict__ W,
                                                 const float* __restrict__ a,
                                                 float* __restrict__ wpack) {
    const int c = threadIdx.x;  // 0..511 (input-feature index)
    float w1 = 0.0f, w2 = 0.0f;
    for (int o = 0; o < F_OUT; ++o) {
        const float w = W[o * F_IN + c];   // W is [out, in] row-major
        w1 += w * a[o];
        w2 += w * a[F_OUT + o];
    }
    float* row = wpack + c * 16;
    row[0] = w1;
    row[1] = w2;
#pragma unroll
    for (int n = 2; n < 16; ++n) row[n] = 0.0f;
}

// ---------------------------------------------------------------------------
// K2: s = nodes @ wpack  via V_WMMA_F32_16X16X4_F32 (full fp32 precision).
// One wave per 16-row tile of `nodes` (512 tiles). K loop of 128 steps.
// A layout (16x4 f32): lane = m(0..15) in low 4 bits; lanes 0-15 carry K={0,1},
// lanes 16-31 carry K={2,3} in VGPR{0,1}  -> contiguous float2 per lane.
// B layout (4x16 f32): symmetric, n = lane&15, same K split.
// D layout (16x16 f32): vgpr r -> m = r + 8*(lane>>4), n = lane&15.
// Only columns n==0 (s1) and n==1 (s2) are stored.
// ---------------------------------------------------------------------------
__global__ __launch_bounds__(256) void gat_scores(const float* __restrict__ nodes,
                                                  const float* __restrict__ wpack,
                                                  float* __restrict__ s1,
                                                  float* __restrict__ s2) {
    const int wave = (blockIdx.x * blockDim.x + threadIdx.x) >> 5;
    const int lane = threadIdx.x & 31;
    const int m0   = wave * 16;
    const int mrow = m0 + (lane & 15);
    const int n    = lane & 15;
    const int koff = (lane >> 4) * 2;     // 0 for lanes 0-15, 2 for lanes 16-31

    const float* arow = nodes + (size_t)mrow * F_IN + koff;
    const float* bcol = wpack + koff * 16 + n;

    v8f acc = {};
    for (int k = 0; k < F_IN; k += 4) {
        v2f av = *(const v2f*)(arow + k);             // A[m][k+koff], A[m][k+koff+1]
        v2f bv;
        bv.x = bcol[k * 16];                          // B[k+koff  ][n]
        bv.y = bcol[k * 16 + 16];                     // B[k+koff+1][n]
        acc = __builtin_amdgcn_wmma_f32_16x16x4_f32(
            /*neg_a=*/false, av, /*neg_b=*/false, bv,
            /*c_mod=*/(short)0, acc, /*reuse_a=*/false, /*reuse_b=*/false);
    }

    if (n < 2) {                                      // lanes 0,1,16,17 carry results
        float* dst = (n == 0) ? s1 : s2;
        const int mbase = m0 + (lane >> 4) * 8;
#pragma unroll
        for (int r = 0; r < 8; ++r) dst[mbase + r] = acc[r];
    }
}

// ---------------------------------------------------------------------------
// K3: per-row masked-softmax streamer. One workgroup (256 threads) per row.
// adj row (32 KB) is read from HBM exactly once (non-temporal), the masked
// leaky-relu logits live in LDS for the max/sum/scale passes.
// ---------------------------------------------------------------------------
__global__ __launch_bounds__(256) void gat_softmax_row(const float* __restrict__ adj,
                                                       const float* __restrict__ s1,
                                                       const float* __restrict__ s2,
                                                       float* __restrict__ out) {
    __shared__ float e_row[N_NODES];   // 32 KB: logits, then probabilities
    __shared__ float red[8];
    __shared__ float bcast;

    const int i    = blockIdx.x;
    const int tid  = threadIdx.x;
    const int lane = tid & 31;
    const int wid  = tid >> 5;
    const float s1i = s1[i];
    const float NEG_INF = -__builtin_inff();
    const float* arow = adj + (size_t)i * N_NODES;

    // ---- pass 1: logits -> LDS, row max ----
    float lmax = NEG_INF;
#pragma unroll
    for (int c = 0; c < N_NODES / (256 * 4); ++c) {
        const int j = (c * 256 + tid) * 4;
        const v4f a4 = __builtin_nontemporal_load((const v4f*)(arow + j));
        const v4f t4 = *(const v4f*)(s2 + j);
#pragma unroll
        for (int r = 0; r < 4; ++r) {
            float x = s1i + t4[r];
            x = (x > 0.0f) ? x : ALPHA * x;            // leaky_relu(0.2)
            const float e = (a4[r] >= 0.5f) ? x : NEG_INF;
            e_row[j + r] = e;
            lmax = fmaxf(lmax, e);
        }
    }
#pragma unroll
    for (int off = 16; off > 0; off >>= 1)
        lmax = fmaxf(lmax, __shfl_xor(lmax, off, 32));
    if (lane == 0) red[wid] = lmax;
    __syncthreads();
    if (tid == 0) {
        float m = red[0];
#pragma unroll
        for (int w = 1; w < 8; ++w) m = fmaxf(m, red[w]);
        bcast = m;
    }
    __syncthreads();
    const float m = bcast;

    // ---- pass 2: exp in place, row sum ----
    float lsum = 0.0f;
#pragma unroll
    for (int c = 0; c < N_NODES / (256 * 4); ++c) {
        const int j = (c * 256 + tid) * 4;
#pragma unroll
        for (int r = 0; r < 4; ++r) {
            const float p = __expf(e_row[j + r] - m);  // exp(-inf)=0 for masked
            e_row[j + r] = p;
            lsum += p;
        }
    }
#pragma unroll
    for (int off = 16; off > 0; off >>= 1)
        lsum += __shfl_xor(lsum, off, 32);
    __syncthreads();                 // red[] reuse
    if (lane == 0) red[wid] = lsum;
    __syncthreads();
    if (tid == 0) {
        float s = red[0];
#pragma unroll
        for (int w = 1; w < 8; ++w) s += red[w];
        bcast = 1.0f / s;
    }
    __syncthreads();
    const float inv = bcast;

    // ---- pass 3: scale + non-temporal store ----
    float* orow = out + (size_t)i * N_NODES;
#pragma unroll
    for (int c = 0; c < N_NODES / (256 * 4); ++c) {
        const int j = (c * 256 + tid) * 4;
        v4f o4;
#pragma unroll
        for (int r = 0; r < 4; ++r) o4[r] = e_row[j + r] * inv;
        __builtin_nontemporal_store(o4, (v4f*)(orow + j));
    }
}

// ---------------------------------------------------------------------------
extern "C" void kernel_launch(void* const* d_in, const int* in_sizes, int n_in,
                              void* d_out, int out_size, void* d_ws, size_t ws_size,
                              hipStream_t stream) {
    const float* nodes = (const float*)d_in[0];  // [8192, 512]
    const float* adj   = (const float*)d_in[1];  // [8192, 8192]
    const float* W     = (const float*)d_in[2];  // [512, 512] (out, in)
    const float* a     = (const float*)d_in[3];  // [1024]
    float* out = (float*)d_out;

    float* wpack = (float*)d_ws;          // 512*16 floats = 32 KB
    float* s1    = wpack + F_IN * 16;     // 8192 floats
    float* s2    = s1 + N_NODES;          // 8192 floats

    gat_wpack<<<1, 512, 0, stream>>>(W, a, wpack);
    gat_scores<<<(N_NODES / 16) / 8, 256, 0, stream>>>(nodes, wpack, s1, s2);
    gat_softmax_row<<<N_NODES, 256, 0, stream>>>(adj, s1, s2, out);
}